// GATModel_55585466745363
// MI455X (gfx1250) — compile-verified
//
#include <hip/hip_runtime.h>
#include <math.h>

// ---------------- problem constants (from reference) ----------------
#define N_NODES  50000     // divisible by 16 -> no GEMM row tail
#define N_EDGES  800000
#define D_FEAT   64
#define HIDDEN   64
#define EDGE_DIM 8
#define LL_DIM   32
#define NEG_SLOPE 0.2f

typedef __attribute__((ext_vector_type(2))) float v2f;
typedef __attribute__((ext_vector_type(8))) float v8f;

// =====================================================================
// WMMA fp32 GEMM:  Y[M x Ncols] = X[M x K] @ W[K x Ncols]
// One wave computes a 16(M) x 64(N) strip with 4 accumulators, stepping
// K by 4 using V_WMMA_F32_16X16X4_F32 (fp32 A/B/C -> matches reference
// precision; GEMMs are a tiny fraction of total work anyway).
//
// Fragment layouts per CDNA5 ISA 7.12.2:
//  A 16x4 f32 : lanes 0-15 = rows, v0 holds K = 2*half, v1 = 2*half+1
//  B 4x16 f32 : lanes 0-15 = cols, v0 holds K = 2*half, v1 = 2*half+1
//  C/D 16x16  : VGPR v -> row v (lanes 0-15) / row v+8 (lanes 16-31)
// =====================================================================
__global__ __launch_bounds__(32)
void gemm_wmma_f32(const float* __restrict__ X, const float* __restrict__ W,
                   float* __restrict__ Y, int K, int Ncols) {
  const int lane = threadIdx.x & 31;
  const int lr   = lane & 15;
  const int half = lane >> 4;
  const int row0 = blockIdx.y << 4;
  const int col0 = blockIdx.x << 6;

  v8f acc0 = {}, acc1 = {}, acc2 = {}, acc3 = {};
  const float* xrow = X + (size_t)(row0 + lr) * K;

  for (int k0 = 0; k0 < K; k0 += 4) {
    const int ka = k0 + 2 * half;
    v2f a;
    a.x = xrow[ka];
    a.y = xrow[ka + 1];
    const float* w0 = W + (size_t)ka * Ncols + col0 + lr;
    const float* w1 = w0 + Ncols;
    v2f b0; b0.x = w0[0];  b0.y = w1[0];
    v2f b1; b1.x = w0[16]; b1.y = w1[16];
    v2f b2; b2.x = w0[32]; b2.y = w1[32];
    v2f b3; b3.x = w0[48]; b3.y = w1[48];
    acc0 = __builtin_amdgcn_wmma_f32_16x16x4_f32(false, a, false, b0, (short)0, acc0, false, false);
    acc1 = __builtin_amdgcn_wmma_f32_16x16x4_f32(false, a, false, b1, (short)0, acc1, false, false);
    acc2 = __builtin_amdgcn_wmma_f32_16x16x4_f32(false, a, false, b2, (short)0, acc2, false, false);
    acc3 = __builtin_amdgcn_wmma_f32_16x16x4_f32(false, a, false, b3, (short)0, acc3, false, false);
  }

  const int orow = row0 + (half << 3);
#pragma unroll
  for (int v = 0; v < 8; ++v) {
    float* yp = Y + (size_t)(orow + v) * Ncols + col0 + lr;
    yp[0]  = acc0[v];
    yp[16] = acc1[v];
    yp[32] = acc2[v];
    yp[48] = acc3[v];
  }
}

// ---------------- per-node attention scalars a_s, a_d ----------------
__global__ void node_att_kernel(const float* __restrict__ xs, const float* __restrict__ xd,
                                const float* __restrict__ attS, const float* __restrict__ attD,
                                float* __restrict__ a_s, float* __restrict__ a_d,
                                int Nn, int H, int C) {
  int n = blockIdx.x * blockDim.x + threadIdx.x;
  if (n >= Nn) return;
  for (int h = 0; h < H; ++h) {
    const float* ps = xs + (size_t)n * H * C + h * C;
    const float* pd = xd + (size_t)n * H * C + h * C;
    float ss = 0.f, sd = 0.f;
    for (int c = 0; c < C; ++c) {
      ss += ps[c] * attS[h * C + c];
      sd += pd[c] * attD[h * C + c];
    }
    a_s[n * H + h] = ss;
    a_d[n * H + h] = sd;
  }
}

// w_eff[k,h] = sum_c W_edge[k, h*C+c] * att_edge[h,c]   (folds edge proj)
__global__ void edge_weff_kernel(const float* __restrict__ Wedge, const float* __restrict__ attE,
                                 float* __restrict__ weff, int H, int C) {
  int i = threadIdx.x;
  if (i >= EDGE_DIM * H) return;
  int k = i / H, h = i % H;
  float s = 0.f;
  for (int c = 0; c < C; ++c) s += Wedge[(size_t)k * H * C + h * C + c] * attE[h * C + c];
  weff[k * H + h] = s;
}

__global__ void init_amax_denom(float* __restrict__ amax, float* __restrict__ denom, int n) {
  int i = blockIdx.x * blockDim.x + threadIdx.x;
  if (i < n) { amax[i] = -INFINITY; denom[i] = 0.f; }
}

__global__ void init_out_bias(float* __restrict__ out, const float* __restrict__ bias,
                              int total, int HC) {
  int i = blockIdx.x * blockDim.x + threadIdx.x;
  if (i < total) out[i] = bias[i & (HC - 1)];  // HC is 128 or 64 (pow2)
}

// float atomic max via ordered-int mapping
__device__ __forceinline__ void atomicMaxF(float* addr, float v) {
  if (v >= 0.f) atomicMax((int*)addr, __float_as_int(v));
  else          atomicMin((unsigned int*)addr, (unsigned int)__float_as_int(v));
}

// pass 1: raw logits + leaky_relu + segment max over dst
__global__ void edge_alpha_kernel(const int* __restrict__ src, const int* __restrict__ dst,
                                  const float* __restrict__ eattr, const float* __restrict__ weff,
                                  const float* __restrict__ a_s, const float* __restrict__ a_d,
                                  float* __restrict__ alpha, float* __restrict__ amax,
                                  int E, int H) {
  int e = blockIdx.x * blockDim.x + threadIdx.x;
  if (e >= E) return;
  int s = src[e], d = dst[e];
  const float4* p = reinterpret_cast<const float4*>(eattr + (size_t)e * EDGE_DIM);
  float4 q0 = p[0], q1 = p[1];
  float ea[EDGE_DIM] = {q0.x, q0.y, q0.z, q0.w, q1.x, q1.y, q1.z, q1.w};
  for (int h = 0; h < H; ++h) {
    float ae = 0.f;
#pragma unroll
    for (int k = 0; k < EDGE_DIM; ++k) ae += ea[k] * weff[k * H + h];
    float t = a_s[s * H + h] + a_d[d * H + h] + ae;
    float al = t > 0.f ? t : NEG_SLOPE * t;
    alpha[(size_t)e * H + h] = al;
    atomicMaxF(&amax[d * H + h], al);
  }
}

// pass 2: ex = exp(alpha - amax[dst]); denom[dst] += ex   (ex overwrites alpha)
__global__ void edge_exp_kernel(const int* __restrict__ dst, const float* __restrict__ amax,
                                float* __restrict__ alpha, float* __restrict__ denom,
                                int E, int H) {
  int i = blockIdx.x * blockDim.x + threadIdx.x;
  if (i >= E * H) return;
  int e = i / H, h = i - e * H;
  int d = dst[e];
  float ex = expf(alpha[i] - amax[d * H + h]);
  alpha[i] = ex;
  atomicAdd(&denom[d * H + h], ex);
}

// pass 3: one wave32 per edge; lane covers HC/32 contiguous channels.
// coalesced gather of xs[src], f32-atomic scatter-add into out[dst]
// (both arrays are 25.6 MB -> resident in MI455X's 192 MB L2).
__global__ void edge_agg_kernel(const int* __restrict__ src, const int* __restrict__ dst,
                                const float* __restrict__ xs, const float* __restrict__ exv,
                                const float* __restrict__ denom, float* __restrict__ out,
                                int E, int H, int C) {
  int tid  = blockIdx.x * blockDim.x + threadIdx.x;
  int e    = tid >> 5;
  int lane = tid & 31;
  if (e >= E) return;
  const int HC  = H * C;
  const int per = HC >> 5;        // 4 (HC=128) or 2 (HC=64); stays within one head
  const int c0  = lane * per;
  const int h   = c0 / C;
  int s = src[e], d = dst[e];
  float w = exv[(size_t)e * H + h] / (denom[d * H + h] + 1e-16f);
  const float* xp = xs + (size_t)s * HC + c0;
  float* op = out + (size_t)d * HC + c0;
#pragma unroll 4
  for (int j = 0; j < per; ++j) atomicAdd(op + j, xp[j] * w);
}

// ---------------- fused head: no nonlinearity between ll and fl ------
__global__ void head_pre_kernel(const float* __restrict__ llW, const float* __restrict__ llb,
                                const float* __restrict__ flW, const float* __restrict__ flb,
                                float* __restrict__ wfin) {
  int i = threadIdx.x;
  if (i < HIDDEN) {
    float s = 0.f;
    for (int j = 0; j < LL_DIM; ++j) s += llW[i * LL_DIM + j] * flW[j];
    wfin[i] = s;
  }
  if (i == 0) {
    float b = flb[0];
    for (int j = 0; j < LL_DIM; ++j) b += llb[j] * flW[j];
    wfin[HIDDEN] = b;
  }
}

__global__ void head_kernel(const float* __restrict__ x, const float* __restrict__ wfin,
                            float* __restrict__ out, int Nn) {
  int n = blockIdx.x * blockDim.x + threadIdx.x;
  if (n >= Nn) return;
  const float* xp = x + (size_t)n * HIDDEN;
  float s = wfin[HIDDEN];
#pragma unroll
  for (int i = 0; i < HIDDEN; ++i) s += xp[i] * wfin[i];
  out[n] = 1.f / (1.f + expf(-s));
}

// =====================================================================
extern "C" void kernel_launch(void* const* d_in, const int* in_sizes, int n_in,
                              void* d_out, int out_size, void* d_ws, size_t ws_size,
                              hipStream_t stream) {
  (void)in_sizes; (void)n_in; (void)out_size; (void)ws_size;

  const float* x0    = (const float*)d_in[0];
  const int*   eidx  = (const int*)d_in[1];
  const float* eattr = (const float*)d_in[2];
  // d_in[3] = batch (unused: single graph, all zeros)
  const int* src = eidx;
  const int* dst = eidx + N_EDGES;

  // params flattened in dict-insertion order: 3 layers x 7 tensors, then head
  const float *Wsrc[3], *Wdst[3], *Wedge[3], *attS[3], *attD[3], *attE[3], *bias[3];
  for (int li = 0; li < 3; ++li) {
    int b = 4 + 7 * li;
    Wsrc[li]  = (const float*)d_in[b + 0];
    Wdst[li]  = (const float*)d_in[b + 1];
    Wedge[li] = (const float*)d_in[b + 2];
    attS[li]  = (const float*)d_in[b + 3];
    attD[li]  = (const float*)d_in[b + 4];
    attE[li]  = (const float*)d_in[b + 5];
    bias[li]  = (const float*)d_in[b + 6];
  }
  const float* llW = (const float*)d_in[25];
  const float* llb = (const float*)d_in[26];
  const float* flW = (const float*)d_in[27];
  const float* flb = (const float*)d_in[28];

  // ---- workspace layout (floats): 3 big slabs + small arrays (~85 MB) ----
  float* ws = (float*)d_ws;
  const size_t NHC = (size_t)N_NODES * 128;
  float* P0    = ws;
  float* P1    = ws + NHC;
  float* P2    = ws + 2 * NHC;
  float* a_s   = ws + 3 * NHC;                     // N*2
  float* a_d   = a_s  + (size_t)N_NODES * 2;       // N*2
  float* alpha = a_d  + (size_t)N_NODES * 2;       // E*2
  float* amax  = alpha + (size_t)N_EDGES * 2;      // N*2
  float* denom = amax + (size_t)N_NODES * 2;       // N*2
  float* weff  = denom + (size_t)N_NODES * 2;      // 16
  float* wfin  = weff + 32;                        // 65

  const int Hs[3] = {2, 2, 1};
  const int Ks[3] = {D_FEAT, 2 * HIDDEN, 2 * HIDDEN};
  // buffer schedule: out reuses the xd slab (xd only needed until a_d is done)
  float* xsbuf[3] = {P0, P0, P0};
  float* xdbuf[3] = {P1, P2, P1};

  const float* xcur = x0;
  for (int li = 0; li < 3; ++li) {
    const int H = Hs[li], C = HIDDEN, HC = H * C, K = Ks[li];
    float* xs  = xsbuf[li];
    float* xd  = xdbuf[li];
    float* out = xd;  // reuse slab

    dim3 ggrid(HC / 64, N_NODES / 16);
    gemm_wmma_f32<<<ggrid, 32, 0, stream>>>(xcur, Wsrc[li], xs, K, HC);
    gemm_wmma_f32<<<ggrid, 32, 0, stream>>>(xcur, Wdst[li], xd, K, HC);

    node_att_kernel<<<(N_NODES + 127) / 128, 128, 0, stream>>>(
        xs, xd, attS[li], attD[li], a_s, a_d, N_NODES, H, C);
    edge_weff_kernel<<<1, 32, 0, stream>>>(Wedge[li], attE[li], weff, H, C);
    init_amax_denom<<<(N_NODES * 2 + 255) / 256, 256, 0, stream>>>(amax, denom, N_NODES * H);

    edge_alpha_kernel<<<(N_EDGES + 255) / 256, 256, 0, stream>>>(
        src, dst, eattr, weff, a_s, a_d, alpha, amax, N_EDGES, H);
    edge_exp_kernel<<<(N_EDGES * H + 255) / 256, 256, 0, stream>>>(
        dst, amax, alpha, denom, N_EDGES, H);

    int total = N_NODES * HC;
    init_out_bias<<<(total + 255) / 256, 256, 0, stream>>>(out, bias[li], total, HC);
    edge_agg_kernel<<<(N_EDGES * 32 + 255) / 256, 256, 0, stream>>>(
        src, dst, xs, alpha, denom, out, N_EDGES, H, C);

    xcur = out;
  }

  head_pre_kernel<<<1, 64, 0, stream>>>(llW, llb, flW, flb, wfin);
  head_kernel<<<(N_NODES + 127) / 128, 128, 0, stream>>>(xcur, wfin, (float*)d_out, N_NODES);
}